// Sggnn_siamese_48000554500540
// MI455X (gfx1250) — compile-verified
//
#include <hip/hip_runtime.h>
#include <hip/hip_bf16.h>

typedef __attribute__((ext_vector_type(2))) float v2f;
typedef __attribute__((ext_vector_type(4))) float v4f;
typedef __attribute__((ext_vector_type(8))) float v8f;
typedef __attribute__((ext_vector_type(4))) unsigned int v4u;
typedef __attribute__((ext_vector_type(4))) int v4i;
typedef __attribute__((ext_vector_type(8))) int v8i;

#define GM 256      // B*N rows
#define GK 24576    // C*H*W
#define GN 1024     // FEAT
#define KC 32       // K chunk staged in LDS per TDM tile

// ---------------------------------------------------------------------------
// TDM descriptor (CDNA5 ISA): DMA a 16(row) x 32(K)
// fp32 tile of X into LDS, padding +1 DWORD per 32-DWORD row so the LDS row
// stride is 33 floats (conflict-free WMMA fragment reads).
// ---------------------------------------------------------------------------
__device__ __forceinline__ void tdm_load_a_tile(unsigned int lds_addr,
                                                unsigned long long gaddr) {
  v4u g0;
  g0[0] = 1u;                                        // count=1 (valid), user mode
  g0[1] = lds_addr;                                  // lds_addr (bytes)
  g0[2] = (unsigned int)(gaddr & 0xFFFFFFFFu);       // global_addr[31:0]
  g0[3] = (unsigned int)((gaddr >> 32) & 0x01FFFFFFu) | (2u << 30);  // ga[56:32] | type=2

  v8i g1;
  // word0: mask=0 | data_size=2(4B)<<16 | pad_enable<<20 | pad_interval=4(32 DW)<<22 | pad_amount=0(1 DW)<<25
  g1[0] = (2 << 16) | (1 << 20) | (4 << 22);
  g1[1] = (int)((GK & 0xFFFF) << 16);                // [31:16]=tensor_dim0[15:0]; barrier addr=0
  g1[2] = (int)((GK >> 16) | ((GM & 0xFFFF) << 16)); // tensor_dim0[31:16] | tensor_dim1[15:0]
  g1[3] = (int)((GM >> 16) | (KC << 16));            // tensor_dim1[31:16] | tile_dim0=32
  g1[4] = 16;                                        // tile_dim1=16 | tile_dim2=0
  g1[5] = GK;                                        // tensor_dim0_stride[31:0]
  g1[6] = (int)(((unsigned long long)GK * GM & 0xFFFFull) << 16); // s0[47:32]=0 | s1[15:0]
  g1[7] = (int)((unsigned long long)GK * GM >> 16);  // tensor_dim1_stride[47:16]

  v4i g2;
  g2[0] = 1;  // tensor_dim2 = 1
  g2[1] = 0;  // tensor_dim3 = 0 (iterate disabled)
  g2[2] = 0;  // tensor_dim2_stride[31:0]
  g2[3] = 0;  // stride[47:32] | tile_dim3=0 (unused)

  v4i g3 = {0, 0, 0, 0};  // tensor_dim3_stride / tensor_dim4 / tile_dim4 unused
  v8i g4 = {0, 0, 0, 0, 0, 0, 0, 0};  // 6-arg builtin's extra group (unused)

  __builtin_amdgcn_tensor_load_to_lds(g0, g1, g2, g3, g4, 0);
}

// ---------------------------------------------------------------------------
// Phase 1: feats(256x1024) = X(256x24576) @ W(24576x1024), exact fp32 WMMA.
// Block = 8 waves; block tile = 16(M) x 128(N); one 16x16 tile per wave.
// A tiles are DMA'd by the TDM into double-buffered LDS while WMMAs run.
// ---------------------------------------------------------------------------
__global__ __launch_bounds__(256) void sggnn_gemm_f32_wmma(
    const float* __restrict__ X, const float* __restrict__ Wm,
    float* __restrict__ feats) {
  __shared__ float As[2][16][KC + 1];  // 33-float row stride (TDM pads 1 DW / row)

  const int tid  = threadIdx.x;
  const int lane = tid & 31;
  const int wave = tid >> 5;
  const int m0   = blockIdx.y * 16;
  const int n0   = blockIdx.x * 128 + wave * 16;

  const int g = lane >> 4;  // lane half -> K offset 2*g
  const int r = lane & 15;  // M row (A frag) / N col (B, C/D frags)

  const float* Bbase = Wm + n0 + r;  // + k*GN per element
  const unsigned long long base_g =
      (unsigned long long)(uintptr_t)(X + (size_t)m0 * GK);
  const unsigned int lds0 = (unsigned int)(uintptr_t)&As[0][0][0];
  const unsigned int lds1 = (unsigned int)(uintptr_t)&As[1][0][0];

  if (wave == 0) tdm_load_a_tile(lds0, base_g);  // prologue: chunk 0 -> buf 0

  v8f acc = {};

  for (int c = 0; c < GK / KC; ++c) {
    const int k0 = c * KC;
    if (wave == 0) __builtin_amdgcn_s_wait_tensorcnt(0);  // chunk c landed
    __syncthreads();  // chunk c visible; all reads of the other buffer done
    if (wave == 0 && k0 + KC < GK)                        // overlap next DMA
      tdm_load_a_tile((c & 1) ? lds0 : lds1,
                      base_g + (unsigned long long)(k0 + KC) * 4ull);

    if (k0 + KC < GK)  // prefetch next B chunk (global_prefetch_b8)
      __builtin_prefetch(Bbase + (size_t)(k0 + KC + 2 * g) * GN, 0, 1);

    const float(*A)[KC + 1] = As[c & 1];
#pragma unroll
    for (int kk = 0; kk < KC; kk += 4) {
      const int ka = kk + 2 * g;  // this lane-half's K base within chunk
      v2f a, b;
      // A 16x4 f32 layout: lane L -> M=L%16, VGPR v -> K = ka + v
      a.x = A[r][ka];
      a.y = A[r][ka + 1];
      // B 4x16 f32 layout (column per lane): lane L -> N=L%16, VGPR v -> K = ka + v
      const size_t krow = (size_t)(k0 + ka) * GN;
      b.x = Bbase[krow];
      b.y = Bbase[krow + GN];
      acc = __builtin_amdgcn_wmma_f32_16x16x4_f32(
          /*neg_a=*/false, a, /*neg_b=*/false, b,
          /*c_mod=*/(short)0, acc, /*reuse_a=*/false, /*reuse_b=*/false);
    }
  }

  // C/D layout: VGPR v holds (M = v + 8*g, N = r)
  float* out = feats + (size_t)(m0 + 8 * g) * GN + n0 + r;
#pragma unroll
  for (int v = 0; v < 8; ++v) out[(size_t)v * GN] = acc[v];
}

// ---------------------------------------------------------------------------
// Phase 2: d[i,j,0,n,k] = (feats[4j, k] - feats[4*((i+j)%64)+1+n, k])^2
// One block per (i,j,n) row; 256 threads x float4 = 1024 k. feats (1MB) stays
// in L2; the 50MB result streams out with non-temporal stores.
// ---------------------------------------------------------------------------
__global__ __launch_bounds__(256) void sggnn_dist(
    const float* __restrict__ feats, float* __restrict__ dout) {
  const int bx = blockIdx.x;  // (i*64 + j)*3 + n, 12288 total
  const int n  = bx % 3;
  const int j  = (bx / 3) & 63;
  const int i  = bx / 192;
  const int k  = threadIdx.x * 4;

  const v4f fp = *(const v4f*)(feats + (size_t)(j * 4) * GN + k);
  const int gi = ((i + j) & 63) * 4 + 1 + n;
  const v4f fg = *(const v4f*)(feats + (size_t)gi * GN + k);

  const v4f df = fp - fg;
  const v4f o  = df * df;
  __builtin_nontemporal_store(o, (v4f*)(dout + (size_t)bx * GN + k));
}

// ---------------------------------------------------------------------------
// Phase 3: w (64x64x3x3) and label (64x64x1x3) equality tables.
// ---------------------------------------------------------------------------
__global__ __launch_bounds__(256) void sggnn_wlabel(
    const int* __restrict__ y, float* __restrict__ out) {
  const size_t D_ELEMS = (size_t)64 * 64 * 3 * 1024;  // 12582912
  const int    W_ELEMS = 64 * 64 * 3 * 3;             // 36864
  const int    L_ELEMS = 64 * 64 * 3;                 // 12288
  const int t = blockIdx.x * blockDim.x + threadIdx.x;

  if (t < W_ELEMS) {
    const int n = t % 3, m = (t / 3) % 3, j = (t / 9) & 63, i = t / 576;
    const int yg  = y[j * 4 + 1 + m];
    const int ygr = y[((i + j) & 63) * 4 + 1 + n];
    out[D_ELEMS + t] = (yg == ygr) ? 1.0f : 0.0f;
  } else if (t < W_ELEMS + L_ELEMS) {
    const int u = t - W_ELEMS;
    const int n = u % 3, j = (u / 3) & 63, i = u / 192;
    const int yp  = y[j * 4];
    const int ygr = y[((i + j) & 63) * 4 + 1 + n];
    out[D_ELEMS + W_ELEMS + u] = (yp == ygr) ? 1.0f : 0.0f;
  }
}

extern "C" void kernel_launch(void* const* d_in, const int* in_sizes, int n_in,
                              void* d_out, int out_size, void* d_ws, size_t ws_size,
                              hipStream_t stream) {
  (void)in_sizes; (void)n_in; (void)out_size; (void)ws_size;
  const float* X  = (const float*)d_in[0];   // x: (64,4,3,128,64) fp32
  const int*   y  = (const int*)d_in[1];     // y: (64,4) int
  const float* Wm = (const float*)d_in[2];   // W: (24576,1024) fp32
  float* feats = (float*)d_ws;               // 256*1024 fp32 = 1 MB scratch
  float* out   = (float*)d_out;

  dim3 gemm_grid(8, 16);  // 8 N-strips x 16 M-tiles = 128 blocks, 1024 waves
  sggnn_gemm_f32_wmma<<<gemm_grid, 256, 0, stream>>>(X, Wm, feats);
  sggnn_dist<<<64 * 64 * 3, 256, 0, stream>>>(feats, out);
  sggnn_wlabel<<<192, 256, 0, stream>>>(y, out);
}